// PeriodicAffineContour_4183298146931
// MI455X (gfx1250) — compile-verified
//
#include <hip/hip_runtime.h>

typedef __attribute__((ext_vector_type(2))) float v2f;
typedef __attribute__((ext_vector_type(8))) float v8f;
typedef __attribute__((ext_vector_type(4))) unsigned int v4u;
typedef __attribute__((ext_vector_type(4))) int v4i;
typedef __attribute__((ext_vector_type(8))) int v8i;

#define FW 128      // MLP hidden width (F)
#define TS 64       // d-rows per TDM tile
#define BD 2048     // d-rows per workgroup

// ---------------------------------------------------------------------------
// Zero the reduction workspace (harness poisons d_ws with 0xAA).
// ---------------------------------------------------------------------------
__global__ __launch_bounds__(256) void pac_init_ws(float* ws) {
    int i = blockIdx.x * blockDim.x + threadIdx.x;
    if (i < 2 * FW + 2) ws[i] = 0.0f;
}

// ---------------------------------------------------------------------------
// TDM: issue tensor_load_to_lds of a [rows x FW] f32 tile (row stride FW)
// from global into LDS at byte offset lds_byte. D# per cdna5 ISA ch.8:
//  g0: count=1 | lds_addr | global_addr | type=2
//  g1: data_size=4B, tensor_dim0=FW, tensor_dim1=rows, tile=FWxrows,
//      tensor_dim0_stride=FW. 2D tile -> groups 2/3 zero.
// ---------------------------------------------------------------------------
__device__ __forceinline__ void tdm_load_tile(const float* gsrc,
                                              unsigned int lds_byte, int rows) {
    unsigned long long ga = (unsigned long long)(uintptr_t)gsrc;
    v4u g0;
    g0.x = 1u;                                         // count=1, user mode
    g0.y = lds_byte;                                   // lds_addr (bytes)
    g0.z = (unsigned int)(ga & 0xFFFFFFFFu);           // global_addr[31:0]
    g0.w = (unsigned int)((ga >> 32) & 0x01FFFFFFu)    // global_addr[56:32]
         | (2u << 30);                                 // type = 2 ("image")
    v8i g1;
    g1[0] = 0x00020000;                                // data_size=2 (4B)
    g1[1] = (int)(FW << 16);                           // tensor_dim0 lo16
    g1[2] = (int)(rows << 16);                         // dim0 hi=0 | tensor_dim1 lo16
    g1[3] = (int)(FW << 16);                           // dim1 hi=0 | tile_dim0
    g1[4] = rows;                                      // tile_dim1, tile_dim2=0
    g1[5] = FW;                                        // tensor_dim0_stride lo32
    g1[6] = 0;
    g1[7] = 0;
    v4i gz = {0, 0, 0, 0};
#if __clang_major__ >= 23
    v8i gz8 = {0, 0, 0, 0, 0, 0, 0, 0};
    __builtin_amdgcn_tensor_load_to_lds(g0, g1, gz, gz, gz8, 0);
#else
    __builtin_amdgcn_tensor_load_to_lds(g0, g1, gz, gz, 0);
#endif
}

// ---------------------------------------------------------------------------
// Phase A: h1_raw[f] = sum_d xs[d]*W1[d][f], h2_raw[f] = sum_d xs[d]*W2[d][f]
// 536 MB weight stream (HBM-bound, ~23 us floor at 23.3 TB/s).
// Double-buffered TDM global->LDS staging; 8 waves x v_wmma_f32_16x16x4_f32
// (A row 0 = xs chunk of 4, B = 4x16 LDS weight tile).
// ---------------------------------------------------------------------------
__global__ __launch_bounds__(256) void pac_phaseA(
        const float* __restrict__ x, const int* __restrict__ even_idx,
        const float* __restrict__ W1, const float* __restrict__ W2,
        float* __restrict__ acc1, float* __restrict__ acc2,
        int Vh, int Dtot)
{
    extern __shared__ float smem[];
    float* xsb = smem;                       // [2][TS] Fourier features
    float* wA  = smem + 2 * TS;              // [2][TS*FW] W1 stages
    float* wB  = wA + 2 * TS * FW;           // [2][TS*FW] W2 stages

    const int t    = threadIdx.x;
    const int lane = t & 31;
    const int wave = t >> 5;                 // 0..7 -> f-block
    const int f0   = wave << 4;
    const int n    = lane & 15;              // N within f-block
    const int kh   = (lane >> 4) << 1;       // lanes 0-15 -> K{0,1}; 16-31 -> K{2,3}
    const int dstart = blockIdx.x * BD;
    const int ntiles = BD / TS;

    v8f c1 = {}; v8f c2 = {};

    // Prime the pipeline: tile 0 into stage 0.
    if (wave == 0) {
        tdm_load_tile(W1 + (size_t)dstart * FW, (unsigned int)(uintptr_t)(wA), TS);
        tdm_load_tile(W2 + (size_t)dstart * FW, (unsigned int)(uintptr_t)(wB), TS);
    }

    for (int tt = 0; tt < ntiles; ++tt) {
        const int stage = tt & 1;
        const int dtile = dstart + tt * TS;
        // Fourier features for this tile (tile lies in one segment: TS | Vh).
        const int seg = dtile / Vh;          // 0:cos(x) 1:sin(x) 2:cos(2x) 3:sin(2x)
        if (t < TS) {
            int i = dtile + t - seg * Vh;
            float a = x[even_idx[i]];
            float ang = (float)((seg >> 1) + 1) * a;
            xsb[stage * TS + t] = (seg & 1) ? __sinf(ang) : __cosf(ang);
        }
        // Issue next tile into the other stage, then wait for current tile.
        if (wave == 0) {
            if (tt + 1 < ntiles) {
                const int ns = (tt + 1) & 1;
                tdm_load_tile(W1 + (size_t)(dtile + TS) * FW,
                              (unsigned int)(uintptr_t)(wA + ns * TS * FW), TS);
                tdm_load_tile(W2 + (size_t)(dtile + TS) * FW,
                              (unsigned int)(uintptr_t)(wB + ns * TS * FW), TS);
                __builtin_amdgcn_s_wait_tensorcnt(2);  // oldest 2 (this tile) done
            } else {
                __builtin_amdgcn_s_wait_tensorcnt(0);
            }
        }
        __syncthreads();

        const float* tA  = wA  + stage * TS * FW;
        const float* tB  = wB  + stage * TS * FW;
        const float* txs = xsb + stage * TS;
        for (int j = 0; j < TS / 4; ++j) {
            const int r = j << 2;
            // Unconditional LDS loads, then value-select (v_cndmask, no EXEC ops).
            float xa = txs[r + kh];
            float xb = txs[r + kh + 1];
            v2f A; A.x = (n == 0) ? xa : 0.0f; A.y = (n == 0) ? xb : 0.0f;
            // B (4x16): lane = N, VGPR0 = K{0|2} row, VGPR1 = K{1|3} row.
            const float* p1 = tA + (r + kh) * FW + f0 + n;
            const float* p2 = tB + (r + kh) * FW + f0 + n;
            v2f B1; B1.x = p1[0]; B1.y = p1[FW];
            v2f B2; B2.x = p2[0]; B2.y = p2[FW];
            c1 = __builtin_amdgcn_wmma_f32_16x16x4_f32(
                     false, A, false, B1, (short)0, c1, false, false);
            c2 = __builtin_amdgcn_wmma_f32_16x16x4_f32(
                     false, A, false, B2, (short)0, c2, false, false);
        }
        __syncthreads();
    }

    // C/D layout: row M=0 lives in VGPR0 on lanes 0-15 (N = lane).
    if (lane < 16) {
        atomicAdd(&acc1[f0 + lane], c1[0]);
        atomicAdd(&acc2[f0 + lane], c2[0]);
    }
    (void)Dtot;
}

// ---------------------------------------------------------------------------
// Phase B: two tiny MLPs (relu -> 128x128 -> relu -> scalar head).
// One block, 256 threads: [0,128) = MLP1 (u_s), [128,256) = MLP2 (u_t).
// ---------------------------------------------------------------------------
__global__ __launch_bounds__(256) void pac_phaseB(
        const float* __restrict__ acc,          // [2][FW] raw first-layer sums
        const float* __restrict__ b1,  const float* __restrict__ Wh1,
        const float* __restrict__ bh1, const float* __restrict__ b2,
        const float* __restrict__ Wh2, const float* __restrict__ bh2,
        const float* __restrict__ Wsv, const float* __restrict__ bs,
        const float* __restrict__ Wtv, const float* __restrict__ bt,
        float* __restrict__ scal)               // [2] -> u_s, u_t
{
    __shared__ float h[2][FW];
    __shared__ float red[2][FW];
    const int t = threadIdx.x;
    const int m = t >> 7;
    const int f = t & (FW - 1);

    const float* bb = m ? b2 : b1;
    h[m][f] = fmaxf(acc[m * FW + f] + bb[f], 0.0f);
    __syncthreads();

    const float* Wh = m ? Wh2 : Wh1;
    const float* bh = m ? bh2 : bh1;
    float s = bh[f];
    for (int j = 0; j < FW; ++j) s = fmaf(h[m][j], Wh[j * FW + f], s);
    s = fmaxf(s, 0.0f);

    const float* wo = m ? Wtv : Wsv;
    red[m][f] = s * wo[f];
    __syncthreads();

    if (f == 0) {
        float r = 0.0f;
        for (int j = 0; j < FW; ++j) r += red[m][j];
        scal[m] = r + (m ? bt[0] : bs[0]);
    }
}

// ---------------------------------------------------------------------------
// Phase C: y = x (complex); y[odd] += i*(u_s*x[odd] + u_t).
// complex64 interleaved: out[2i] = re, out[2i+1] = im.
// ---------------------------------------------------------------------------
__global__ __launch_bounds__(256) void pac_phaseC(
        const float* __restrict__ x, const float* __restrict__ scal,
        float* __restrict__ out, int V)
{
    int i = blockIdx.x * blockDim.x + threadIdx.x;
    if (i >= V) return;
    float xi = x[i];
    float us = scal[0];
    float ut = scal[1];
    out[2 * i]     = xi;
    out[2 * i + 1] = (i & 1) ? fmaf(us, xi, ut) : 0.0f;
}

// ---------------------------------------------------------------------------
extern "C" void kernel_launch(void* const* d_in, const int* in_sizes, int n_in,
                              void* d_out, int out_size, void* d_ws, size_t ws_size,
                              hipStream_t stream) {
    const float* x   = (const float*)d_in[0];
    const int*   ev  = (const int*)d_in[1];
    const int*   od  = (const int*)d_in[2];   (void)od;
    const float* W1  = (const float*)d_in[3];
    const float* b1  = (const float*)d_in[4];
    const float* Wh1 = (const float*)d_in[5];
    const float* bh1 = (const float*)d_in[6];
    const float* W2  = (const float*)d_in[7];
    const float* b2  = (const float*)d_in[8];
    const float* Wh2 = (const float*)d_in[9];
    const float* bh2 = (const float*)d_in[10];
    const float* Wsv = (const float*)d_in[11];
    const float* bs  = (const float*)d_in[12];
    const float* Wtv = (const float*)d_in[13];
    const float* bt  = (const float*)d_in[14];

    const int V  = in_sizes[0];
    const int Vh = in_sizes[1];
    const int D  = in_sizes[3] / FW;   // W1 is [D, FW]

    float* ws   = (float*)d_ws;
    float* acc1 = ws;            // [FW]
    float* acc2 = ws + FW;       // [FW]
    float* scal = ws + 2 * FW;   // [2]
    float* out  = (float*)d_out;
    (void)n_in; (void)out_size; (void)ws_size;

    const size_t smem_bytes = (size_t)(2 * TS + 2 * 2 * TS * FW) * sizeof(float);

    pac_init_ws<<<2, 256, 0, stream>>>(ws);
    pac_phaseA<<<D / BD, 256, smem_bytes, stream>>>(x, ev, W1, W2, acc1, acc2, Vh, D);
    pac_phaseB<<<1, 256, 0, stream>>>(ws, b1, Wh1, bh1, b2, Wh2, bh2,
                                      Wsv, bs, Wtv, bt, scal);
    pac_phaseC<<<(V + 255) / 256, 256, 0, stream>>>(x, scal, out, V);
}